// rotation_loss_34368328302906
// MI455X (gfx1250) — compile-verified
//
#include <hip/hip_runtime.h>
#include <math.h>

typedef __attribute__((ext_vector_type(2))) float v2f;
typedef __attribute__((ext_vector_type(8))) float v8f;

#define TPB 256
#define MAX_BLOCKS 2048
#define EPS 1e-12f

// ---------------------------------------------------------------------------
// Per-element math (matches the JAX reference exactly)
// ---------------------------------------------------------------------------
__device__ __forceinline__ void quat_from_mat(const float r[9], float q[4]) {
    const float r00 = r[0], r01 = r[1], r02 = r[2];
    const float r10 = r[3], r11 = r[4], r12 = r[5];
    const float r20 = r[6], r21 = r[7], r22 = r[8];

    const float tr = r00 + r11 + r22;
    const float S0 = sqrtf(fmaxf(tr + 1.0f, EPS)) * 2.0f;
    const float S1 = sqrtf(fmaxf(1.0f + r00 - r11 - r22, EPS)) * 2.0f;
    const float S2 = sqrtf(fmaxf(1.0f + r11 - r00 - r22, EPS)) * 2.0f;
    const float S3 = sqrtf(fmaxf(1.0f + r22 - r00 - r11, EPS)) * 2.0f;

    const float q0w = 0.25f * S0,      q0x = (r21 - r12) / S0, q0y = (r02 - r20) / S0, q0z = (r10 - r01) / S0;
    const float q1w = (r21 - r12) / S1, q1x = 0.25f * S1,      q1y = (r10 + r01) / S1, q1z = (r02 + r20) / S1;
    const float q2w = (r02 - r20) / S2, q2x = (r10 + r01) / S2, q2y = 0.25f * S2,      q2z = (r21 + r12) / S2;
    const float q3w = (r10 - r01) / S3, q3x = (r02 + r20) / S3, q3y = (r21 + r12) / S3, q3z = 0.25f * S3;

    const bool c0 = (tr > 0.0f);
    const bool c1 = (r00 > r11) && (r00 > r22);
    const bool c2 = (r11 > r22);

    // Branchless nested select (keeps the streaming loop convergent)
    float qw = c0 ? q0w : (c1 ? q1w : (c2 ? q2w : q3w));
    float qx = c0 ? q0x : (c1 ? q1x : (c2 ? q2x : q3x));
    float qy = c0 ? q0y : (c1 ? q1y : (c2 ? q2y : q3y));
    float qz = c0 ? q0z : (c1 ? q1z : (c2 ? q2z : q3z));

    const float n = sqrtf(qw * qw + qx * qx + qy * qy + qz * qz);
    q[0] = qw / n; q[1] = qx / n; q[2] = qy / n; q[3] = qz / n;
}

__device__ __forceinline__ float quat_distance(const float q[4], const float r[4]) {
    // rinv = conj(r) / |r|^2
    const float s  = r[0] * r[0] + r[1] * r[1] + r[2] * r[2] + r[3] * r[3];
    const float w2 =  r[0] / s, x2 = -r[1] / s, y2 = -r[2] / s, z2 = -r[3] / s;
    const float w1 = q[0], x1 = q[1], y1 = q[2], z1 = q[3];

    const float t0 = w2 * w1 - x2 * x1 - y2 * y1 - z2 * z1;
    const float t1 = w2 * x1 + x2 * w1 - y2 * z1 + z2 * y1;
    const float t2 = w2 * y1 + x2 * z1 + y2 * w1 - z2 * x1;
    const float t3 = w2 * z1 - x2 * y1 + y2 * x1 + z2 * w1;

    return 2.0f * atan2f(sqrtf(t1 * t1 + t2 * t2 + t3 * t3), fabsf(t0));
}

// ---------------------------------------------------------------------------
// Wave32 sum-reduction through the matrix pipe:
//   A (16x4 f32): a0 = s(lane), a1 = 0  ->  A[m][0]=s(m), A[m][2]=s(m+16)
//   B = ones     ->  D[m][n] = s(m) + s(m+16)
//   Sum 8 C/D VGPRs per lane: lanes 0-15 get s(0..7)+s(16..23),
//                             lanes 16-31 get s(8..15)+s(24..31)
//   One shfl_xor(16) finishes the 32-lane sum (exact f32 math throughout).
// ---------------------------------------------------------------------------
__device__ __forceinline__ float wave32_sum_wmma(float s) {
    v2f a; a[0] = s;     a[1] = 0.0f;
    v2f b; b[0] = 1.0f;  b[1] = 1.0f;
    v8f c = {0.0f, 0.0f, 0.0f, 0.0f, 0.0f, 0.0f, 0.0f, 0.0f};
    v8f d = __builtin_amdgcn_wmma_f32_16x16x4_f32(
        /*neg_a=*/false, a, /*neg_b=*/false, b,
        /*c_mod=*/(short)0, c, /*reuse_a=*/false, /*reuse_b=*/false);
    float t = d[0] + d[1] + d[2] + d[3] + d[4] + d[5] + d[6] + d[7];
    t += __shfl_xor(t, 16, 32);
    return t;
}

// ---------------------------------------------------------------------------
// Kernel 1: streaming per-block partial sums (bandwidth-bound main loop)
// ---------------------------------------------------------------------------
__global__ void rotation_loss_partial(const float* __restrict__ P,
                                      const float* __restrict__ G,
                                      float* __restrict__ part, int n) {
    float acc = 0.0f;
    const int stride = gridDim.x * blockDim.x;
    for (int i = blockIdx.x * blockDim.x + threadIdx.x; i < n; i += stride) {
        float rp[9], rg[9];
        const float* pp = P + (size_t)i * 9;
        const float* pg = G + (size_t)i * 9;
#pragma unroll
        for (int k = 0; k < 9; ++k) rp[k] = pp[k];
#pragma unroll
        for (int k = 0; k < 9; ++k) rg[k] = pg[k];

        float qp[4], qg[4];
        quat_from_mat(rp, qp);
        quat_from_mat(rg, qg);
        acc += quat_distance(qp, qg);
    }

    // EXEC is all-ones here (loop has reconverged) -> WMMA is legal.
    const float wsum = wave32_sum_wmma(acc);

    __shared__ float warp_part[TPB / 32];
    const int lane = threadIdx.x & 31;
    const int wid  = threadIdx.x >> 5;
    if (lane == 0) warp_part[wid] = wsum;
    __syncthreads();
    if (threadIdx.x == 0) {
        float bsum = 0.0f;
#pragma unroll
        for (int w = 0; w < TPB / 32; ++w) bsum += warp_part[w];
        part[blockIdx.x] = bsum;
    }
}

// ---------------------------------------------------------------------------
// Kernel 2: deterministic single-block reduction of block partials -> mean
// ---------------------------------------------------------------------------
__global__ void rotation_loss_final(const float* __restrict__ part, int nparts,
                                    float* __restrict__ out, float inv_n) {
    __shared__ float sm[TPB];
    float s = 0.0f;
    for (int i = threadIdx.x; i < nparts; i += TPB) s += part[i];
    sm[threadIdx.x] = s;
    __syncthreads();
#pragma unroll
    for (int off = TPB / 2; off > 0; off >>= 1) {
        if (threadIdx.x < off) sm[threadIdx.x] += sm[threadIdx.x + off];
        __syncthreads();
    }
    if (threadIdx.x == 0) out[0] = sm[0] * inv_n;
}

// ---------------------------------------------------------------------------
extern "C" void kernel_launch(void* const* d_in, const int* in_sizes, int n_in,
                              void* d_out, int out_size, void* d_ws, size_t ws_size,
                              hipStream_t stream) {
    const float* pred = (const float*)d_in[0];   // (B,3,3) f32
    const float* gt   = (const float*)d_in[1];   // (B,3,3) f32
    float* out        = (float*)d_out;           // scalar f32
    float* part       = (float*)d_ws;            // block partials

    const int n = in_sizes[0] / 9;               // number of matrices

    int nblocks = (n + TPB - 1) / TPB;
    if (nblocks > MAX_BLOCKS) nblocks = MAX_BLOCKS;
    const int ws_cap = (int)(ws_size / sizeof(float));
    if (nblocks > ws_cap) nblocks = ws_cap;
    if (nblocks < 1) nblocks = 1;

    rotation_loss_partial<<<nblocks, TPB, 0, stream>>>(pred, gt, part, n);
    rotation_loss_final<<<1, TPB, 0, stream>>>(part, nblocks, out, 1.0f / (float)n);
}